// LambdaLayer1D_4698694222460
// MI455X (gfx1250) — compile-verified
//
#include <hip/hip_runtime.h>
#include <math.h>

typedef __attribute__((ext_vector_type(2))) float v2f;
typedef __attribute__((ext_vector_type(8))) float v8f;

constexpr int Bb = 8, Cc = 512, Tt = 4096;
constexpr int Hh = 4, Kk = 16, Rr = 15, Vv = 128;
constexpr int QKV_CH = 64 + 16 + 128; // 208 rows: q[0:64), k[64:80), v[80:208)
constexpr float EPSf = 1e-5f;

// ---------------------------------------------------------------------------
// Kernel 1: qkv projections.  Per batch: OUT[208,4096] = W[208,512] * X[512,4096]
// One wave owns a 16(M)x64(N) strip: 4 independent WMMA accumulator chains
// (latency hiding) and 4x reuse of each A (weight) fragment.
// A layout (16x4 f32): lane&15 = M; lanes<16 hold K={0,1}, lanes>=16 hold K={2,3}.
// B layout (4x16 f32): lane&15 = N; lanes<16 hold K rows {0,1}, lanes>=16 {2,3}.
// D layout (16x16 f32): lane&15 = N; vgpr i = row (i + 8*(lane>>4)).
// ---------------------------------------------------------------------------
__global__ __launch_bounds__(128) void proj_gemm(
    const float* __restrict__ x, const float* __restrict__ Wq,
    const float* __restrict__ Wk, const float* __restrict__ Wv,
    float* __restrict__ qkv)
{
    const int b     = blockIdx.z;
    const int mrow0 = blockIdx.y * 16;          // 13 M-tiles (208 rows)
    const int wave  = threadIdx.x >> 5;
    const int lane  = threadIdx.x & 31;
    const int hi    = lane >> 4;
    const int lo    = lane & 15;
    const int t0    = blockIdx.x * 256 + wave * 64;

    const float* W; int wr;
    if (mrow0 < 64)      { W = Wq; wr = mrow0;      }
    else if (mrow0 < 80) { W = Wk; wr = mrow0 - 64; }
    else                 { W = Wv; wr = mrow0 - 80; }

    const float* __restrict__ Arow = W + (size_t)(wr + lo) * Cc + 2 * hi;
    const float* __restrict__ Xb   =
        x + (size_t)b * Cc * Tt + (size_t)(2 * hi) * Tt + (t0 + lo);

    v8f acc0 = {}, acc1 = {}, acc2 = {}, acc3 = {};
    #pragma unroll 4
    for (int k0 = 0; k0 < Cc; k0 += 4) {
        v2f a = *(const v2f*)(Arow + k0);        // contiguous, 8B aligned
        const float* __restrict__ xp = Xb + (size_t)k0 * Tt;
        v2f b0, b1, b2, b3;
        b0.x = xp[ 0]; b0.y = xp[(size_t)Tt +  0];
        b1.x = xp[16]; b1.y = xp[(size_t)Tt + 16];
        b2.x = xp[32]; b2.y = xp[(size_t)Tt + 32];
        b3.x = xp[48]; b3.y = xp[(size_t)Tt + 48];
        acc0 = __builtin_amdgcn_wmma_f32_16x16x4_f32(false, a, false, b0, (short)0, acc0, false, false);
        acc1 = __builtin_amdgcn_wmma_f32_16x16x4_f32(false, a, false, b1, (short)0, acc1, false, false);
        acc2 = __builtin_amdgcn_wmma_f32_16x16x4_f32(false, a, false, b2, (short)0, acc2, false, false);
        acc3 = __builtin_amdgcn_wmma_f32_16x16x4_f32(false, a, false, b3, (short)0, acc3, false, false);
    }

    float* __restrict__ O = qkv + ((size_t)b * QKV_CH + mrow0) * Tt + (t0 + lo);
    #pragma unroll
    for (int i = 0; i < 8; ++i) {
        size_t row = (size_t)(i + 8 * hi) * Tt;
        O[row +  0] = acc0[i];
        O[row + 16] = acc1[i];
        O[row + 32] = acc2[i];
        O[row + 48] = acc3[i];
    }
}

// ---------------------------------------------------------------------------
// Kernel 2: BatchNorm training stats per channel over (B, T) = 32768 samples.
// ---------------------------------------------------------------------------
__global__ __launch_bounds__(256) void bn_stats(
    const float* __restrict__ qkv, int chan0,
    float* __restrict__ mean, float* __restrict__ rstd)
{
    __shared__ float s1[256], s2[256];
    const int ch = blockIdx.x, tid = threadIdx.x;
    float s = 0.f, sq = 0.f;
    for (int i = tid; i < Bb * Tt; i += 256) {
        int b = i >> 12, t = i & (Tt - 1);
        float v = qkv[((size_t)b * QKV_CH + chan0 + ch) * Tt + t];
        s += v; sq += v * v;
    }
    s1[tid] = s; s2[tid] = sq;
    __syncthreads();
    for (int off = 128; off > 0; off >>= 1) {
        if (tid < off) { s1[tid] += s1[tid + off]; s2[tid] += s2[tid + off]; }
        __syncthreads();
    }
    if (tid == 0) {
        float m   = s1[0] * (1.f / (Bb * Tt));
        float var = s2[0] * (1.f / (Bb * Tt)) - m * m;
        mean[ch] = m;
        rstd[ch] = rsqrtf(var + EPSf);
    }
}

// ---------------------------------------------------------------------------
// Kernel 3: softmax stats (row max + 1/sum(exp)) per (b, k-channel) over T.
// ---------------------------------------------------------------------------
__global__ __launch_bounds__(256) void softmax_stats(
    const float* __restrict__ qkv,
    float* __restrict__ kmax, float* __restrict__ kinv)
{
    __shared__ float sh[256];
    const int bk = blockIdx.x;                   // [0, 128)
    const int b = bk >> 4, kc = bk & 15;
    const int tid = threadIdx.x;
    const float* row = qkv + ((size_t)b * QKV_CH + 64 + kc) * Tt;

    float m = -3.4e38f;
    for (int t = tid; t < Tt; t += 256) m = fmaxf(m, row[t]);
    sh[tid] = m; __syncthreads();
    for (int off = 128; off > 0; off >>= 1) {
        if (tid < off) sh[tid] = fmaxf(sh[tid], sh[tid + off]);
        __syncthreads();
    }
    float M = sh[0];
    __syncthreads();

    float s = 0.f;
    for (int t = tid; t < Tt; t += 256) s += expf(row[t] - M);
    sh[tid] = s; __syncthreads();
    for (int off = 128; off > 0; off >>= 1) {
        if (tid < off) sh[tid] += sh[tid + off];
        __syncthreads();
    }
    if (tid == 0) { kmax[bk] = M; kinv[bk] = 1.f / sh[0]; }
}

// ---------------------------------------------------------------------------
// Kernel 4: apply BN (q, v channels) and softmax (k channels) elementwise.
// ---------------------------------------------------------------------------
__global__ __launch_bounds__(256) void normalize(
    const float* __restrict__ raw, float* __restrict__ outn,
    const float* __restrict__ mean_q, const float* __restrict__ rstd_q,
    const float* __restrict__ gamma_q, const float* __restrict__ beta_q,
    const float* __restrict__ mean_v, const float* __restrict__ rstd_v,
    const float* __restrict__ gamma_v, const float* __restrict__ beta_v,
    const float* __restrict__ kmax, const float* __restrict__ kinv)
{
    size_t idx = (size_t)blockIdx.x * blockDim.x + threadIdx.x;
    const size_t n = (size_t)Bb * QKV_CH * Tt;
    if (idx >= n) return;
    int bc = (int)(idx / Tt);
    int ch = bc % QKV_CH;
    int b  = bc / QKV_CH;
    float xv = raw[idx], y;
    if (ch < 64) {
        y = (xv - mean_q[ch]) * rstd_q[ch] * gamma_q[ch] + beta_q[ch];
    } else if (ch < 80) {
        int kc = ch - 64;
        y = expf(xv - kmax[b * 16 + kc]) * kinv[b * 16 + kc];
    } else {
        int vc = ch - 80;
        y = (xv - mean_v[vc]) * rstd_v[vc] * gamma_v[vc] + beta_v[vc];
    }
    outn[idx] = y;
}

// ---------------------------------------------------------------------------
// Kernel 5: lambda_c[b,16,128] = kn[16,T] * vn[T,128] via WMMA over t.
// Two interleaved accumulator chains to break the serial D->C dependency;
// both A and B per-lane loads are contiguous float2's along t.
// ---------------------------------------------------------------------------
__global__ __launch_bounds__(128) void lambdac_gemm(
    const float* __restrict__ qn, float* __restrict__ lc)
{
    const int b    = blockIdx.y;
    const int wave = threadIdx.x >> 5;
    const int lane = threadIdx.x & 31;
    const int hi = lane >> 4, lo = lane & 15;
    const int n0 = (blockIdx.x * 4 + wave) * 16;   // v-tile

    const float* __restrict__ Arow = qn + ((size_t)b * QKV_CH + 64 + lo) * Tt + 2 * hi;
    const float* __restrict__ Brow = qn + ((size_t)b * QKV_CH + 80 + n0 + lo) * Tt + 2 * hi;

    v8f acc0 = {}, acc1 = {};
    #pragma unroll 4
    for (int tk = 0; tk < Tt; tk += 8) {
        v2f a0 = *(const v2f*)(Arow + tk);
        v2f b0 = *(const v2f*)(Brow + tk);
        v2f a1 = *(const v2f*)(Arow + tk + 4);
        v2f b1 = *(const v2f*)(Brow + tk + 4);
        acc0 = __builtin_amdgcn_wmma_f32_16x16x4_f32(false, a0, false, b0, (short)0, acc0, false, false);
        acc1 = __builtin_amdgcn_wmma_f32_16x16x4_f32(false, a1, false, b1, (short)0, acc1, false, false);
    }
    acc0 = acc0 + acc1;

    float* __restrict__ O = lc + (size_t)b * Kk * Vv + n0 + lo;
    #pragma unroll
    for (int i = 0; i < 8; ++i)
        O[(size_t)(i + 8 * hi) * Vv] = acc0[i];
}

// ---------------------------------------------------------------------------
// Kernel 6: fused position-conv + both lambda contractions.
// y[b,h,v,t] = sum_k q[b,h*16+k,t] * (lambda_c[k,v] + pos_b[k] + conv_k(vn[v, t-7..t+7]))
// LDS stages vn (with halo), q tile, lambda_c (+pos_b folded), conv weights.
// All 64 q values for this thread's t are hoisted into registers once.
// ---------------------------------------------------------------------------
__global__ __launch_bounds__(256) void final_fused(
    const float* __restrict__ qn, const float* __restrict__ lc,
    const float* __restrict__ pos_w, const float* __restrict__ pos_b,
    float* __restrict__ out)
{
    constexpr int TT = 64, HALO = 7;
    constexpr int WCOLS = TT + 2 * HALO;   // 78
    constexpr int WSTR  = WCOLS + 1;       // 79 (pad)
    __shared__ float sVn[Vv * WSTR];       // ~40 KB
    __shared__ float sQ [64 * TT];         // 16 KB
    __shared__ float sLc[Kk * Vv];         // 8 KB
    __shared__ float sW [Kk * Rr];

    const int b  = blockIdx.y;
    const int t0 = blockIdx.x * TT;
    const int tid = threadIdx.x;
    const float* qb = qn + (size_t)b * QKV_CH * Tt;

    for (int i = tid; i < Vv * WCOLS; i += 256) {
        int v = i / WCOLS, col = i % WCOLS;
        int t = t0 - HALO + col;
        sVn[v * WSTR + col] = (t >= 0 && t < Tt) ? qb[(size_t)(80 + v) * Tt + t] : 0.f;
    }
    for (int i = tid; i < 64 * TT; i += 256) {
        int ch = i / TT, col = i % TT;
        sQ[i] = qb[(size_t)ch * Tt + t0 + col];
    }
    for (int i = tid; i < Kk * Vv; i += 256)
        sLc[i] = lc[(size_t)b * Kk * Vv + i] + pos_b[i / Vv];
    for (int i = tid; i < Kk * Rr; i += 256) sW[i] = pos_w[i];
    __syncthreads();

    const int tloc = tid & (TT - 1);
    const int vg   = tid >> 6;             // 0..3

    // Hoist q[h*16+k, tloc] for all 64 (h,k) into registers (v-invariant).
    float qreg[64];
    #pragma unroll
    for (int i = 0; i < 64; ++i) qreg[i] = sQ[i * TT + tloc];

    float* ob = out + (size_t)b * (Hh * Vv) * Tt + t0 + tloc;

    for (int vi = 0; vi < Vv / 4; ++vi) {
        int v = vg + vi * 4;
        float win[Rr];
        #pragma unroll
        for (int r = 0; r < Rr; ++r) win[r] = sVn[v * WSTR + tloc + r];
        float a0 = 0.f, a1 = 0.f, a2 = 0.f, a3 = 0.f;
        #pragma unroll
        for (int k = 0; k < Kk; ++k) {
            float lp = sLc[k * Vv + v];
            #pragma unroll
            for (int r = 0; r < Rr; ++r) lp += sW[k * Rr + r] * win[r];
            a0 += qreg[ 0 + k] * lp;
            a1 += qreg[16 + k] * lp;
            a2 += qreg[32 + k] * lp;
            a3 += qreg[48 + k] * lp;
        }
        ob[(size_t)(0 * Vv + v) * Tt] = a0;
        ob[(size_t)(1 * Vv + v) * Tt] = a1;
        ob[(size_t)(2 * Vv + v) * Tt] = a2;
        ob[(size_t)(3 * Vv + v) * Tt] = a3;
    }
}

// ---------------------------------------------------------------------------
extern "C" void kernel_launch(void* const* d_in, const int* in_sizes, int n_in,
                              void* d_out, int out_size, void* d_ws, size_t ws_size,
                              hipStream_t stream)
{
    (void)in_sizes; (void)n_in; (void)out_size; (void)ws_size;
    const float* x       = (const float*)d_in[0];
    const float* Wq      = (const float*)d_in[1];
    const float* Wk      = (const float*)d_in[2];
    const float* Wv      = (const float*)d_in[3];
    const float* gamma_q = (const float*)d_in[4];
    const float* beta_q  = (const float*)d_in[5];
    const float* gamma_v = (const float*)d_in[6];
    const float* beta_v  = (const float*)d_in[7];
    const float* pos_w   = (const float*)d_in[8];
    const float* pos_b   = (const float*)d_in[9];
    float* out = (float*)d_out;

    const size_t elems_qkv = (size_t)Bb * QKV_CH * Tt;   // 6.8M floats
    float* qkv_raw = (float*)d_ws;
    float* qkv_n   = qkv_raw + elems_qkv;
    float* mean_q  = qkv_n + elems_qkv;
    float* rstd_q  = mean_q + 64;
    float* mean_v  = rstd_q + 64;
    float* rstd_v  = mean_v + 128;
    float* kmaxp   = rstd_v + 128;
    float* kinvp   = kmaxp + 128;
    float* lcbuf   = kinvp + 128;                        // 8*16*128

    proj_gemm<<<dim3(Tt / 256, 13, Bb), 128, 0, stream>>>(x, Wq, Wk, Wv, qkv_raw);
    bn_stats<<<64, 256, 0, stream>>>(qkv_raw, 0, mean_q, rstd_q);
    bn_stats<<<128, 256, 0, stream>>>(qkv_raw, 80, mean_v, rstd_v);
    softmax_stats<<<128, 256, 0, stream>>>(qkv_raw, kmaxp, kinvp);
    int nblk = (int)((elems_qkv + 255) / 256);
    normalize<<<nblk, 256, 0, stream>>>(qkv_raw, qkv_n,
                                        mean_q, rstd_q, gamma_q, beta_q,
                                        mean_v, rstd_v, gamma_v, beta_v,
                                        kmaxp, kinvp);
    lambdac_gemm<<<dim3(2, Bb), 128, 0, stream>>>(qkv_n, lcbuf);
    final_fused<<<dim3(Tt / 64, Bb), 256, 0, stream>>>(qkv_n, lcbuf, pos_w, pos_b, out);
}